// ProjectBEV_70308614636282
// MI455X (gfx1250) — compile-verified
//
#include <hip/hip_runtime.h>
#include <hip/hip_bf16.h>
#include <math.h>

// ---- Problem constants (fixed by the reference) ----
static constexpr unsigned B_DIM = 16;
static constexpr unsigned H_DIM = 64;
static constexpr unsigned W_DIM = 2048;
static constexpr unsigned CK    = 3;
static constexpr unsigned HW    = H_DIM * W_DIM;           // 131072 = 1 << 17
static constexpr size_t   NTOT  = (size_t)B_DIM * CK * HW; // 6,291,456

// Output layout inside d_out (floats), concatenated in reference return order:
//   [0   , 6N )  presence_indices  [N,6]
//   [6N  , 7N )  presence_vals     [N]
//   [7N  , 21N)  position_indices  [2N,7]
//   [21N , 23N)  position_vals     [2N]
static constexpr size_t OFF_PIDX = 0;
static constexpr size_t OFF_PVAL = 6 * NTOT;
static constexpr size_t OFF_QIDX = 7 * NTOT;
static constexpr size_t OFF_QVAL = 21 * NTOT;

#define THREADS 256
#define TILE_FLOATS (THREADS * CK * 2)          // 1536 floats = 6144 bytes
#define TILE_BYTES  (TILE_FLOATS * 4)

typedef __attribute__((ext_vector_type(2))) float v2f;

// Pointer types matching the builtin's parameter type from the diagnostic:
//   '__attribute__((__vector_size__(4 * sizeof(int)))) int *'
typedef int v4i_vs __attribute__((vector_size(4 * sizeof(int))));
typedef __attribute__((address_space(3))) v4i_vs lds_v4i_vs;

__device__ __forceinline__ v2f mk2(float a, float b) { v2f r; r.x = a; r.y = b; return r; }

// Non-temporal stores: the 579MB write stream must not pollute L2 (192MB).
__device__ __forceinline__ void st_nt(v2f v, float* p) {
  __builtin_nontemporal_store(v, (v2f*)p);
}
__device__ __forceinline__ void st_nt1(float v, float* p) {
  __builtin_nontemporal_store(v, p);
}

#if defined(__has_builtin)
#if __has_builtin(__builtin_amdgcn_global_load_async_to_lds_b128)
#define HAVE_ASYNC_LDS 1
#endif
#endif

__global__ __launch_bounds__(THREADS)
void ProjectBEV_scatter_kernel(const float* __restrict__ in, float* __restrict__ out) {
  const unsigned tid = threadIdx.x;
  const unsigned bx  = blockIdx.x;   // hw tile index: 0..HW/THREADS-1
  const unsigned b   = blockIdx.y;   // batch: 0..15

  // Contiguous 6144B input tile for this block: elements (b, hw in [bx*256, +256), c, d)
  const float* gsrc = in + ((size_t)b * HW + (size_t)bx * THREADS) * (CK * 2);

#if HAVE_ASYNC_LDS
  __shared__ float tile[TILE_FLOATS];
  {
    // 384 x b128 lane-transfers: 256 threads do one, first 128 do a second.
    v4i_vs*     gp0 = (v4i_vs*)((const char*)gsrc + (size_t)tid * 16u);
    lds_v4i_vs* lp0 = (lds_v4i_vs*)((char*)tile + (size_t)tid * 16u);
    __builtin_amdgcn_global_load_async_to_lds_b128(gp0, lp0, 0, 0);
    if (tid < (TILE_BYTES - THREADS * 16) / 16) {
      v4i_vs*     gp1 = (v4i_vs*)((const char*)gsrc + (size_t)(THREADS * 16u) + (size_t)tid * 16u);
      lds_v4i_vs* lp1 = (lds_v4i_vs*)((char*)tile + (size_t)(THREADS * 16u) + (size_t)tid * 16u);
      __builtin_amdgcn_global_load_async_to_lds_b128(gp1, lp1, 0, 0);
    }
  }
#if __has_builtin(__builtin_amdgcn_s_wait_asynccnt)
  __builtin_amdgcn_s_wait_asynccnt(0);
#else
  asm volatile("s_wait_asynccnt 0" ::: "memory");
#endif
  __syncthreads();
  const float* src = tile;   // LDS reads: stride-6-dword float2, bank-conflict-free
#else
  const float* src = gsrc;   // fallback: direct global reads (24B per thread)
#endif

  const unsigned hw = bx * THREADS + tid;
  const unsigned w  = hw & (W_DIM - 1u);
  const unsigned h  = hw >> 11;            // W = 2^11
  const float fb = (float)b;
  const float fh = (float)h;
  const float fw = (float)w;
  const size_t nbase = ((size_t)(b * CK) << 17) + hw;  // n for c=0; +c<<17 per channel

  float* __restrict__ pidx = out + OFF_PIDX;
  float* __restrict__ pval = out + OFF_PVAL;
  float* __restrict__ qidx = out + OFF_QIDX;
  float* __restrict__ qval = out + OFF_QVAL;

#pragma unroll
  for (unsigned c = 0; c < CK; ++c) {
    const float x  = src[(size_t)tid * 6u + c * 2u + 0u];
    const float y  = src[(size_t)tid * 6u + c * 2u + 1u];
    const float sx = x * 2.0f + 1.0f;    // SCALE, X_OFF
    const float sy = y * 2.0f + 81.0f;   // SCALE, Y_OFF
    const float qx = truncf(sx);         // == (float)(int64)sx for these magnitudes
    const float qy = truncf(sy);
    const float fc = (float)c;
    const size_t n = nbase + ((size_t)c << 17);

    // presence_indices[n] = [b, qx, qy, c, h, w]   (24B, 8B-aligned, per-thread contiguous)
    float* p = pidx + 6u * n;
    st_nt(mk2(fb, qx), p + 0);
    st_nt(mk2(qy, fc), p + 2);
    st_nt(mk2(fh, fw), p + 4);

    // presence_vals[n] = 1
    st_nt1(1.0f, pval + n);

    // position_indices[2n],[2n+1] = [b,qx,qy,c,h,w,0],[b,qx,qy,c,h,w,1]  (56B contiguous)
    float* q = qidx + 14u * n;
    st_nt(mk2(fb, qx), q + 0);
    st_nt(mk2(qy, fc), q + 2);
    st_nt(mk2(fh, fw), q + 4);
    st_nt(mk2(0.0f, fb), q + 6);
    st_nt(mk2(qx, qy), q + 8);
    st_nt(mk2(fc, fh), q + 10);
    st_nt(mk2(fw, 1.0f), q + 12);

    // position_vals[2n..2n+1] = [sx, sy]
    st_nt(mk2(sx, sy), qval + 2u * n);
  }
}

extern "C" void kernel_launch(void* const* d_in, const int* in_sizes, int n_in,
                              void* d_out, int out_size, void* d_ws, size_t ws_size,
                              hipStream_t stream) {
  (void)in_sizes; (void)n_in; (void)out_size; (void)d_ws; (void)ws_size;
  const float* in = (const float*)d_in[0];
  float* out = (float*)d_out;
  dim3 grid(HW / THREADS, B_DIM, 1);   // 512 x 16 blocks, 256 threads each
  ProjectBEV_scatter_kernel<<<grid, dim3(THREADS, 1, 1), 0, stream>>>(in, out);
}